// H2GCN_87342454931707
// MI455X (gfx1250) — compile-verified
//
#include <hip/hip_runtime.h>
#include <math.h>

#define NNODES 50000
#define INCH   256
#define HIDC   128
#define OUTC   64
#define E1N    800000
#define E2N    1600000
#define BN_EPS 1e-5f

typedef __attribute__((ext_vector_type(2))) float v2f;
typedef __attribute__((ext_vector_type(8))) float v8f;

__device__ __forceinline__ v8f wmma4(v2f a, v2f b, v8f c) {
    // D = A(16x4 f32) * B(4x16 f32) + C(16x16 f32)
    return __builtin_amdgcn_wmma_f32_16x16x4_f32(
        /*neg_a=*/false, a, /*neg_b=*/false, b,
        /*c_mod=*/(short)0, c, /*reuse_a=*/false, /*reuse_b=*/false);
}

// ---------------------------------------------------------------- utilities
__global__ void k_zero_i32(int* __restrict__ p, int n) {
    int i = blockIdx.x * blockDim.x + threadIdx.x;
    for (; i < n; i += gridDim.x * blockDim.x) p[i] = 0;
}

__global__ void k_count_deg(const int* __restrict__ row, int E, int* __restrict__ degp1) {
    int e = blockIdx.x * blockDim.x + threadIdx.x;
    for (; e < E; e += gridDim.x * blockDim.x) atomicAdd(&degp1[row[e] + 1], 1);
}

// inclusive scan, 1024 elems per block (256 thr x 4)
__global__ __launch_bounds__(256) void k_scan_blocks(int* __restrict__ data, int n,
                                                     int* __restrict__ sums) {
    __shared__ int sh[256];
    int tid = threadIdx.x;
    int base = blockIdx.x * 1024 + tid * 4;
    int v[4];
#pragma unroll
    for (int j = 0; j < 4; j++) v[j] = (base + j < n) ? data[base + j] : 0;
    int s = v[0] + v[1] + v[2] + v[3];
    sh[tid] = s;
    __syncthreads();
    for (int off = 1; off < 256; off <<= 1) {
        int t = (tid >= off) ? sh[tid - off] : 0;
        __syncthreads();
        sh[tid] += t;
        __syncthreads();
    }
    int run = sh[tid] - s;  // exclusive prefix for this thread
#pragma unroll
    for (int j = 0; j < 4; j++) {
        run += v[j];
        if (base + j < n) data[base + j] = run;
    }
    if (tid == 255) sums[blockIdx.x] = sh[255];
}

__global__ void k_scan_sums(int* __restrict__ sums, int nb) {
    if (threadIdx.x == 0 && blockIdx.x == 0) {
        int run = 0;
        for (int i = 0; i < nb; i++) { int t = sums[i]; sums[i] = run; run += t; }
    }
}

__global__ __launch_bounds__(256) void k_scan_add(int* __restrict__ data, int n,
                                                  const int* __restrict__ sums) {
    int add = sums[blockIdx.x];
    int base = blockIdx.x * 1024 + threadIdx.x * 4;
#pragma unroll
    for (int j = 0; j < 4; j++)
        if (base + j < n) data[base + j] += add;
}

__global__ void k_scatter(const int* __restrict__ row, const int* __restrict__ col,
                          const float* __restrict__ val, int E,
                          const int* __restrict__ rp, int* __restrict__ cur,
                          int* __restrict__ ec, float* __restrict__ ev) {
    int e = blockIdx.x * blockDim.x + threadIdx.x;
    for (; e < E; e += gridDim.x * blockDim.x) {
        int r = row[e];
        int p = rp[r] + atomicAdd(&cur[r], 1);
        ec[p] = col[e];
        ev[p] = val[e];
    }
}

// ------------------------------------------------ h0 = relu(x @ W + b), WMMA
__global__ __launch_bounds__(256) void k_embed(const float* __restrict__ x,
                                               const float* __restrict__ W,
                                               const float* __restrict__ bias,
                                               float* __restrict__ h0) {
    __shared__ float tile[16 * 260];  // 16 rows x 256 cols, pad 4 to kill bank conflicts
    int tid = threadIdx.x;
    int m0 = blockIdx.x * 16;
    // Stage the x-tile in LDS (compiler-visible stores so the ds_loads of the
    // A-operand stay live and software-pipeline between the WMMAs).
    for (int idx = tid; idx < 1024; idx += 256) {
        int r = idx >> 6, c4 = (idx & 63) << 2;
        float4 v = *(const float4*)(x + (size_t)(m0 + r) * INCH + c4);
        *(float4*)(&tile[r * 260 + c4]) = v;
    }
    __syncthreads();

    int wave = tid >> 5, lane = tid & 31, hi = lane >> 4, ln = lane & 15;
    int n = wave * 16 + ln;  // 8 waves cover the 128 output columns
    v8f acc = {};
    for (int k = 0; k < INCH; k += 4) {
        int kk = k + 2 * hi;
        v2f a; a.x = tile[ln * 260 + kk]; a.y = tile[ln * 260 + kk + 1];
        v2f b; b.x = W[(size_t)kk * HIDC + n]; b.y = W[(size_t)(kk + 1) * HIDC + n];
        acc = wmma4(a, b, acc);
    }
    float bv = bias[n];
#pragma unroll
    for (int r = 0; r < 8; r++) {
        float vv = acc[r] + bv;
        vv = vv > 0.f ? vv : 0.f;
        h0[(size_t)(m0 + r + 8 * hi) * HIDC + n] = vv;
    }
}

// --------------------------- h1 = BN(cat(A1 h0, A2 h0)), one wave per row
// Edge metadata is fetched 32-at-a-time lane-parallel, rows are prefetched,
// then broadcast via shfl so the b128 gathers are independent and pipeline.
__device__ __forceinline__ float4 spmm_row_128(const float* __restrict__ src,
                                               const int* __restrict__ ec,
                                               const float* __restrict__ ev,
                                               int e0, int e1, int lane, int j) {
    float4 acc = make_float4(0.f, 0.f, 0.f, 0.f);
    for (int eb = e0; eb < e1; eb += 32) {
        int nn = e1 - eb; if (nn > 32) nn = 32;
        int cl = 0; float vl = 0.f;
        if (lane < nn) {
            cl = ec[eb + lane];
            vl = ev[eb + lane];
            __builtin_prefetch(src + (size_t)cl * HIDC, 0, 3);  // global_prefetch_b8
        }
        for (int i = 0; i < nn; i++) {
            int   c = __shfl(cl, i);
            float v = __shfl(vl, i);
            float4 t = *(const float4*)(src + (size_t)c * HIDC + j);
            acc.x += v * t.x; acc.y += v * t.y; acc.z += v * t.z; acc.w += v * t.w;
        }
    }
    return acc;
}

__global__ __launch_bounds__(256) void k_spmm1_bn(
    const float* __restrict__ h0,
    const int* __restrict__ rp1, const int* __restrict__ ec1, const float* __restrict__ ev1,
    const int* __restrict__ rp2, const int* __restrict__ ec2, const float* __restrict__ ev2,
    const float* __restrict__ gamma, const float* __restrict__ beta,
    const float* __restrict__ mean, const float* __restrict__ var,
    float* __restrict__ h1) {
    int lane = threadIdx.x & 31;
    int row = (blockIdx.x * blockDim.x + threadIdx.x) >> 5;
    if (row >= NNODES) return;
    int j = lane * 4;

    for (int half = 0; half < 2; half++) {
        const int*   rp = half ? rp2 : rp1;
        const int*   ec = half ? ec2 : ec1;
        const float* ev = half ? ev2 : ev1;
        float4 acc = spmm_row_128(h0, ec, ev, rp[row], rp[row + 1], lane, j);
        int jc = half * HIDC + j;
        float4 g = *(const float4*)(gamma + jc), bb = *(const float4*)(beta + jc);
        float4 mm = *(const float4*)(mean + jc), vv = *(const float4*)(var + jc);
        float4 o;
        o.x = g.x * (acc.x - mm.x) * rsqrtf(vv.x + BN_EPS) + bb.x;
        o.y = g.y * (acc.y - mm.y) * rsqrtf(vv.y + BN_EPS) + bb.y;
        o.z = g.z * (acc.z - mm.z) * rsqrtf(vv.z + BN_EPS) + bb.z;
        o.w = g.w * (acc.w - mm.w) * rsqrtf(vv.w + BN_EPS) + bb.w;
        *(float4*)(h1 + (size_t)row * (2 * HIDC) + jc) = o;
    }
}

// --------------------------- h2 = cat(A1 h1, A2 h1), 256-wide input rows
__global__ __launch_bounds__(256) void k_spmm2(
    const float* __restrict__ h1,
    const int* __restrict__ rp1, const int* __restrict__ ec1, const float* __restrict__ ev1,
    const int* __restrict__ rp2, const int* __restrict__ ec2, const float* __restrict__ ev2,
    float* __restrict__ h2) {
    int lane = threadIdx.x & 31;
    int row = (blockIdx.x * blockDim.x + threadIdx.x) >> 5;
    if (row >= NNODES) return;
    int ja = lane * 4, jb = HIDC + lane * 4;  // two float4 per lane cover 256 cols

    for (int half = 0; half < 2; half++) {
        const int*   rp = half ? rp2 : rp1;
        const int*   ec = half ? ec2 : ec1;
        const float* ev = half ? ev2 : ev1;
        int e0 = rp[row], e1 = rp[row + 1];
        float4 a0 = make_float4(0.f, 0.f, 0.f, 0.f), a1 = a0;
        for (int eb = e0; eb < e1; eb += 32) {
            int nn = e1 - eb; if (nn > 32) nn = 32;
            int cl = 0; float vl = 0.f;
            if (lane < nn) {
                cl = ec[eb + lane];
                vl = ev[eb + lane];
                __builtin_prefetch(h1 + (size_t)cl * (2 * HIDC), 0, 3);
                __builtin_prefetch(h1 + (size_t)cl * (2 * HIDC) + HIDC, 0, 3);
            }
            for (int i = 0; i < nn; i++) {
                int   c = __shfl(cl, i);
                float v = __shfl(vl, i);
                const float* base = h1 + (size_t)c * (2 * HIDC);
                float4 t0 = *(const float4*)(base + ja);
                float4 t1 = *(const float4*)(base + jb);
                a0.x += v * t0.x; a0.y += v * t0.y; a0.z += v * t0.z; a0.w += v * t0.w;
                a1.x += v * t1.x; a1.y += v * t1.y; a1.z += v * t1.z; a1.w += v * t1.w;
            }
        }
        float* o = h2 + (size_t)row * (4 * HIDC) + half * (2 * HIDC);
        *(float4*)(o + ja) = a0;
        *(float4*)(o + jb) = a1;
    }
}

// --------- out = normalize_rows(cat(h0,h1,h2) @ Wf + bf), WMMA + shfl reduce
__device__ __forceinline__ void gemm_span(const float* __restrict__ arow, int kelems,
                                          const float* __restrict__ wbase,
                                          int ln, int hi,
                                          v8f& c0, v8f& c1, v8f& c2, v8f& c3) {
    for (int k = 0; k < kelems; k += 4) {
        int kk = k + 2 * hi;
        float2 a2 = *(const float2*)(arow + kk);
        v2f a; a.x = a2.x; a.y = a2.y;
        const float* w0 = wbase + (size_t)kk * OUTC;
        v2f b;
        b.x = w0[ln];      b.y = w0[OUTC + ln];      c0 = wmma4(a, b, c0);
        b.x = w0[16 + ln]; b.y = w0[OUTC + 16 + ln]; c1 = wmma4(a, b, c1);
        b.x = w0[32 + ln]; b.y = w0[OUTC + 32 + ln]; c2 = wmma4(a, b, c2);
        b.x = w0[48 + ln]; b.y = w0[OUTC + 48 + ln]; c3 = wmma4(a, b, c3);
    }
}

__global__ __launch_bounds__(256) void k_final(const float* __restrict__ h0,
                                               const float* __restrict__ h1,
                                               const float* __restrict__ h2,
                                               const float* __restrict__ Wf,
                                               const float* __restrict__ bf,
                                               float* __restrict__ out) {
    int wid = blockIdx.x * 8 + (threadIdx.x >> 5);
    if (wid >= NNODES / 16) return;
    int lane = threadIdx.x & 31, hi = lane >> 4, ln = lane & 15;
    int m0 = wid * 16;
    int m = m0 + ln;  // A-operand row for this lane

    v8f c0 = {}, c1 = {}, c2 = {}, c3 = {};
    // JK-cat: branch-free spans (boundaries 128/384 are multiples of the K-step)
    gemm_span(h0 + (size_t)m * HIDC,       HIDC,     Wf,                       ln, hi, c0, c1, c2, c3);
    gemm_span(h1 + (size_t)m * (2 * HIDC), 2 * HIDC, Wf + (size_t)128 * OUTC,  ln, hi, c0, c1, c2, c3);
    gemm_span(h2 + (size_t)m * (4 * HIDC), 4 * HIDC, Wf + (size_t)384 * OUTC,  ln, hi, c0, c1, c2, c3);

    float b0 = bf[ln], b1 = bf[16 + ln], b2 = bf[32 + ln], b3 = bf[48 + ln];
    float inv[8];
#pragma unroll
    for (int r = 0; r < 8; r++) {
        c0[r] += b0; c1[r] += b1; c2[r] += b2; c3[r] += b3;
        float p = c0[r] * c0[r] + c1[r] * c1[r] + c2[r] * c2[r] + c3[r] * c3[r];
        p += __shfl_xor(p, 1);
        p += __shfl_xor(p, 2);
        p += __shfl_xor(p, 4);
        p += __shfl_xor(p, 8);  // masks < 16: halves (rows r vs 8+r) stay separate
        float nrm = sqrtf(p);
        inv[r] = 1.0f / fmaxf(nrm, 1e-12f);
    }
#pragma unroll
    for (int r = 0; r < 8; r++) {
        float* o = out + (size_t)(m0 + r + 8 * hi) * OUTC;
        o[ln]      = c0[r] * inv[r];
        o[16 + ln] = c1[r] * inv[r];
        o[32 + ln] = c2[r] * inv[r];
        o[48 + ln] = c3[r] * inv[r];
    }
}

// ---------------------------------------------------------------- launcher
extern "C" void kernel_launch(void* const* d_in, const int* in_sizes, int n_in,
                              void* d_out, int out_size, void* d_ws, size_t ws_size,
                              hipStream_t stream) {
    const float* x       = (const float*)d_in[0];
    const int*   row1    = (const int*)d_in[1];
    const int*   col1    = (const int*)d_in[2];
    const float* val1    = (const float*)d_in[3];
    const int*   row2    = (const int*)d_in[4];
    const int*   col2    = (const int*)d_in[5];
    const float* val2    = (const float*)d_in[6];
    const float* W_embed = (const float*)d_in[7];
    const float* b_embed = (const float*)d_in[8];
    const float* gamma   = (const float*)d_in[9];
    const float* beta    = (const float*)d_in[10];
    const float* bn_mean = (const float*)d_in[11];
    const float* bn_var  = (const float*)d_in[12];
    const float* Wf      = (const float*)d_in[13];
    const float* bf      = (const float*)d_in[14];
    float*       out     = (float*)d_out;

    char* ws = (char*)d_ws;
    size_t off = 0;
    auto take = [&](size_t bytes) -> char* {
        char* p = ws + off;
        off = (off + bytes + 255) & ~(size_t)255;
        return p;
    };
    float* h0  = (float*)take((size_t)NNODES * HIDC * 4);
    float* h1  = (float*)take((size_t)NNODES * 2 * HIDC * 4);
    float* h2  = (float*)take((size_t)NNODES * 4 * HIDC * 4);
    int* rp1   = (int*)take((size_t)(NNODES + 1) * 4);
    int* rp2   = (int*)take((size_t)(NNODES + 1) * 4);
    int* cur1  = (int*)take((size_t)NNODES * 4);
    int* cur2  = (int*)take((size_t)NNODES * 4);
    int* ec1   = (int*)take((size_t)E1N * 4);
    float* ev1 = (float*)take((size_t)E1N * 4);
    int* ec2   = (int*)take((size_t)E2N * 4);
    float* ev2 = (float*)take((size_t)E2N * 4);
    int* sums1 = (int*)take(256);
    int* sums2 = (int*)take(256);

    const int NB = (NNODES + 1 + 1023) / 1024;  // scan blocks (49)

    // CSR build for both adjacency lists
    k_zero_i32<<<196, 256, 0, stream>>>(rp1, NNODES + 1);
    k_zero_i32<<<196, 256, 0, stream>>>(rp2, NNODES + 1);
    k_zero_i32<<<196, 256, 0, stream>>>(cur1, NNODES);
    k_zero_i32<<<196, 256, 0, stream>>>(cur2, NNODES);
    k_count_deg<<<1024, 256, 0, stream>>>(row1, E1N, rp1);
    k_count_deg<<<1024, 256, 0, stream>>>(row2, E2N, rp2);
    k_scan_blocks<<<NB, 256, 0, stream>>>(rp1, NNODES + 1, sums1);
    k_scan_blocks<<<NB, 256, 0, stream>>>(rp2, NNODES + 1, sums2);
    k_scan_sums<<<1, 32, 0, stream>>>(sums1, NB);
    k_scan_sums<<<1, 32, 0, stream>>>(sums2, NB);
    k_scan_add<<<NB, 256, 0, stream>>>(rp1, NNODES + 1, sums1);
    k_scan_add<<<NB, 256, 0, stream>>>(rp2, NNODES + 1, sums2);
    k_scatter<<<1024, 256, 0, stream>>>(row1, col1, val1, E1N, rp1, cur1, ec1, ev1);
    k_scatter<<<1024, 256, 0, stream>>>(row2, col2, val2, E2N, rp2, cur2, ec2, ev2);

    // dense embed (independent of the CSR data; same stream, runs after it)
    k_embed<<<NNODES / 16, 256, 0, stream>>>(x, W_embed, b_embed, h0);

    // two SpMM rounds (L2-resident gathers), then fused projection + L2 norm
    k_spmm1_bn<<<NNODES / 8, 256, 0, stream>>>(h0, rp1, ec1, ev1, rp2, ec2, ev2,
                                               gamma, beta, bn_mean, bn_var, h1);
    k_spmm2<<<NNODES / 8, 256, 0, stream>>>(h1, rp1, ec1, ev1, rp2, ec2, ev2, h2);
    k_final<<<(NNODES / 16 + 7) / 8, 256, 0, stream>>>(h0, h1, h2, Wf, bf, out);
}